// SpatialTransformer_28716151341659
// MI455X (gfx1250) — compile-verified
//
#include <hip/hip_runtime.h>

// Reference constants: B=1, C=32, H=W=512, zero-pad=1 per spatial side.
#define HDIM 512
#define WDIM 512
#define CDIM 32
#define HW   (HDIM * WDIM)

typedef float f2 __attribute__((ext_vector_type(2)));

// Per-pixel tap info (padded-coordinate clamping exactly as the reference).
struct Tap {
    int   ia, ib, ic, id;     // flat (H*W) offsets of the 4 taps (unpadded coords)
    float wa, wb, wc, wd;     // bilinear weights
    bool  va, vb, vc, vd;     // tap validity (false -> tap is in the zero pad ring)
    bool  all;                // all four taps in-bounds
};

__device__ __forceinline__ Tap make_tap(float xf, float yf) {
    const int fx = (int)floorf(xf);
    const int fy = (int)floorf(yf);
    // Clamp in padded coords [0, 513] like the reference.
    const int x0 = min(max(fx,     0), WDIM + 1);
    const int x1 = min(max(fx + 1, 0), WDIM + 1);
    const int y0 = min(max(fy,     0), HDIM + 1);
    const int y1 = min(max(fy + 1, 0), HDIM + 1);

    // Weights from the *clamped* x1/y1 (reference edge behavior).
    const float dxw = (float)x1 - xf;
    const float dyw = (float)y1 - yf;

    Tap t;
    t.wa = dxw * dyw;
    t.wb = dxw * (1.0f - dyw);
    t.wc = (1.0f - dxw) * dyw;
    t.wd = (1.0f - dxw) * (1.0f - dyw);

    const int ax0 = x0 - 1, ax1 = x1 - 1, ay0 = y0 - 1, ay1 = y1 - 1;
    const bool vx0 = (unsigned)ax0 < (unsigned)WDIM;
    const bool vx1 = (unsigned)ax1 < (unsigned)WDIM;
    const bool vy0 = (unsigned)ay0 < (unsigned)HDIM;
    const bool vy1 = (unsigned)ay1 < (unsigned)HDIM;
    const int r0 = ay0 * WDIM;
    const int r1 = ay1 * WDIM;

    t.ia = r0 + ax0; t.ib = r0 + ax1;
    t.ic = r1 + ax0; t.id = r1 + ax1;
    t.va = vy0 && vx0; t.vb = vy0 && vx1;
    t.vc = vy1 && vx0; t.vd = vy1 && vx1;
    t.all = t.va & t.vb & t.vc & t.vd;
    return t;
}

// Two adjacent pixels per thread: b64 NT flow loads, b64 NT output stores,
// scalar b32 gathers (image stays RT-cached; whole 33.5MB image is resident
// in the 192MB L2, so gathers run at L2 bandwidth).
__global__ __launch_bounds__(256) void st_bilinear_warp2(
    const float* __restrict__ img,   // (C,H,W)
    const float* __restrict__ dxT,   // (H,W)
    const float* __restrict__ dyT,   // (H,W)
    float* __restrict__ out)         // (C,H,W), flipped along H
{
    const int t = blockIdx.x * 256 + threadIdx.x;   // pixel-pair id
    const int p = t << 1;                           // even pixel id
    const int h = p >> 9;                           // W == 512
    const int w = p & (WDIM - 1);                   // even, pair stays in row

    // Streamed (read-once) flow: aligned 8B non-temporal loads.
    const f2 dx2 = __builtin_nontemporal_load((const f2*)(dxT + p));
    const f2 dy2 = __builtin_nontemporal_load((const f2*)(dyT + p));

    const Tap t0 = make_tap(dx2.x + (float)(w + 1), dy2.x + (float)(h + 1));
    const Tap t1 = make_tap(dx2.y + (float)(w + 2), dy2.y + (float)(h + 1));

    // Flip(axis=H) folded into the store address; (w even) => 8B aligned.
    float* o = out + (HDIM - 1 - h) * WDIM + w;

    if (t0.all & t1.all) {
        // Fast path (vast majority of waves): 8 unpredicated gathers/channel.
        __builtin_prefetch(img + t0.ia, 0, 3);   // global_prefetch_b8
        __builtin_prefetch(img + t0.ic, 0, 3);
#pragma unroll 4
        for (int c = 0; c < CDIM; ++c) {
            const int off = c * HW;
            const float a0 = img[off + t0.ia], b0 = img[off + t0.ib];
            const float c0 = img[off + t0.ic], d0 = img[off + t0.id];
            const float a1 = img[off + t1.ia], b1 = img[off + t1.ib];
            const float c1 = img[off + t1.ic], d1 = img[off + t1.id];
            f2 v;
            v.x = t0.wa * a0 + t0.wb * b0 + t0.wc * c0 + t0.wd * d0;
            v.y = t1.wa * a1 + t1.wb * b1 + t1.wc * c1 + t1.wd * d1;
            __builtin_nontemporal_store(v, (f2*)(o + off));  // write-once
        }
    } else {
        // Border path: per-tap zero fill (reference's zero pad ring).
#pragma unroll 2
        for (int c = 0; c < CDIM; ++c) {
            const int off = c * HW;
            const float a0 = t0.va ? img[off + t0.ia] : 0.0f;
            const float b0 = t0.vb ? img[off + t0.ib] : 0.0f;
            const float c0 = t0.vc ? img[off + t0.ic] : 0.0f;
            const float d0 = t0.vd ? img[off + t0.id] : 0.0f;
            const float a1 = t1.va ? img[off + t1.ia] : 0.0f;
            const float b1 = t1.vb ? img[off + t1.ib] : 0.0f;
            const float c1 = t1.vc ? img[off + t1.ic] : 0.0f;
            const float d1 = t1.vd ? img[off + t1.id] : 0.0f;
            f2 v;
            v.x = t0.wa * a0 + t0.wb * b0 + t0.wc * c0 + t0.wd * d0;
            v.y = t1.wa * a1 + t1.wb * b1 + t1.wc * c1 + t1.wd * d1;
            __builtin_nontemporal_store(v, (f2*)(o + off));
        }
    }
}

extern "C" void kernel_launch(void* const* d_in, const int* in_sizes, int n_in,
                              void* d_out, int out_size, void* d_ws, size_t ws_size,
                              hipStream_t stream) {
    (void)in_sizes; (void)n_in; (void)d_ws; (void)ws_size; (void)out_size;
    const float* I   = (const float*)d_in[0];   // (1,32,512,512) f32
    const float* dxT = (const float*)d_in[1];   // (1,512,512)    f32
    const float* dyT = (const float*)d_in[2];   // (1,512,512)    f32
    float* out = (float*)d_out;                 // (1,32,512,512) f32

    const int pairs = (HDIM * WDIM) / 2;        // 131072 threads
    st_bilinear_warp2<<<pairs / 256, 256, 0, stream>>>(I, dxT, dyT, out);
}